// FieldAwareFM_85633057947693
// MI455X (gfx1250) — compile-verified
//
#include <hip/hip_runtime.h>
#include <hip/hip_bf16.h>
#include <stdint.h>

// ---------------- FieldAwareFM problem constants ----------------
#define NUM_FIELDS 10
#define FACTOR_DIM 8
#define INPUT_DIM  188610
#define OFF8       188500   // field 8 offset (dim 100)
#define OFF9       188600   // field 9 offset (dim 10)
#define DIM8       100
#define DIM9       10

// ---------------- CDNA5 TDM (tensor_load_to_lds) ----------------
typedef unsigned int v4u __attribute__((ext_vector_type(4)));
typedef int          v4i __attribute__((ext_vector_type(4)));
typedef int          v8i __attribute__((ext_vector_type(8)));

// 2-D tile load, 4-byte elements: tile_d0 contiguous elements per line,
// tile_d1 lines, line stride = stride0 elements. Descriptor packed per
// CDNA5 ISA ch.8 (D# group0 = 128b, group1 = 256b; groups 2/3 unused for
// <=2D tensors). This toolchain's builtin takes 6 args (extra int32x8).
__device__ __forceinline__ void tdm_load_2d(uint32_t lds_byte, const void* gaddr,
                                            uint32_t tile_d0, uint32_t tile_d1,
                                            uint64_t stride0_elems) {
  uint64_t ga = (uint64_t)(uintptr_t)gaddr;
  v4u g0;
  g0[0] = 1u;                                              // count=1, user desc
  g0[1] = lds_byte;                                        // lds_addr [63:32]
  g0[2] = (uint32_t)ga;                                    // global_addr[31:0]
  g0[3] = (uint32_t)((ga >> 32) & 0x01FFFFFFull)           // global_addr[56:32]
        | (2u << 30);                                      // type=2 ("image")
  v8i g1;
  g1[0] = (int)(2u << 16);                                 // data_size = 4B
  g1[1] = (int)((tile_d0 & 0xFFFFu) << 16);                // tensor_dim0[15:0]
  g1[2] = (int)(((tile_d0 >> 16) & 0xFFFFu)                // tensor_dim0[31:16]
              | ((tile_d1 & 0xFFFFu) << 16));              // tensor_dim1[15:0]
  g1[3] = (int)(((tile_d1 >> 16) & 0xFFFFu)                // tensor_dim1[31:16]
              | ((tile_d0 & 0xFFFFu) << 16));              // tile_dim0
  g1[4] = (int)(tile_d1 & 0xFFFFu);                        // tile_dim1 (tile_dim2=0)
  g1[5] = (int)(uint32_t)(stride0_elems & 0xFFFFFFFFull);  // dim0_stride[31:0]
  g1[6] = (int)(uint32_t)((stride0_elems >> 32) & 0xFFFFull); // dim0_stride[47:32]
  g1[7] = 0;
  v4i z4 = {0, 0, 0, 0};
  v8i z8 = {0, 0, 0, 0, 0, 0, 0, 0};
  __builtin_amdgcn_tensor_load_to_lds(g0, g1, z4, z4, z8, 0);
}

// ---------------- compute helpers ----------------
__device__ __forceinline__ float dot8(float4 a0, float4 a1, float4 b0, float4 b1) {
  float s = a0.x * b0.x;
  s = fmaf(a0.y, b0.y, s);
  s = fmaf(a0.z, b0.z, s);
  s = fmaf(a0.w, b0.w, s);
  s = fmaf(a1.x, b1.x, s);
  s = fmaf(a1.y, b1.y, s);
  s = fmaf(a1.z, b1.z, s);
  s = fmaf(a1.w, b1.w, s);
  return s;
}

__device__ __forceinline__ void load_row(const float* p, float4& lo, float4& hi) {
  lo = ((const float4*)p)[0];   // 32B rows are 16B aligned -> 2x b128
  hi = ((const float4*)p)[1];
}

// ---------------- kernel ----------------
__global__ __launch_bounds__(256) void FieldAwareFM_85633057947693_kernel(
    const int* __restrict__ x, const float* __restrict__ wlin,
    const float* __restrict__ blin, const float* __restrict__ emb,
    float* __restrict__ out, int batch) {
  // Hot sub-tables: rows in field-8 / field-9 index ranges of ALL 10 tables.
  __shared__ __align__(16) float s8[NUM_FIELDS * DIM8 * FACTOR_DIM]; // 32.0 KB
  __shared__ __align__(16) float s9[NUM_FIELDS * DIM9 * FACTOR_DIM]; //  3.2 KB

  if (threadIdx.x < 32) {  // wave 0 drives the TDM DMA for the block
    tdm_load_2d((uint32_t)(uintptr_t)s8, emb + (size_t)OFF8 * FACTOR_DIM,
                DIM8 * FACTOR_DIM, NUM_FIELDS, (uint64_t)INPUT_DIM * FACTOR_DIM);
    tdm_load_2d((uint32_t)(uintptr_t)s9, emb + (size_t)OFF9 * FACTOR_DIM,
                DIM9 * FACTOR_DIM, NUM_FIELDS, (uint64_t)INPUT_DIM * FACTOR_DIM);
    __builtin_amdgcn_s_wait_tensorcnt(0);
  }
  __syncthreads();

  const int b = blockIdx.x * blockDim.x + threadIdx.x;
  if (b >= batch) return;

  static constexpr int kOff[NUM_FIELDS] = {0, 100000, 150000, 170000, 180000,
                                           185000, 187000, 188000, OFF8, OFF9};
  // 10 ints per row, rows are 8B aligned -> five b64 loads
  int raw[NUM_FIELDS];
  const int2* xp = (const int2*)(x + (size_t)b * NUM_FIELDS);
  #pragma unroll
  for (int i = 0; i < NUM_FIELDS / 2; ++i) {
    int2 v = xp[i];
    raw[2 * i]     = v.x;
    raw[2 * i + 1] = v.y;
  }
  int xo[NUM_FIELDS];
  #pragma unroll
  for (int f = 0; f < NUM_FIELDS; ++f) xo[f] = raw[f] + kOff[f];

  // linear term
  float acc = blin[0];
  #pragma unroll
  for (int f = 0; f < NUM_FIELDS; ++f) acc += wlin[xo[f]];

  // pairwise field-aware interactions: sum_{f<g} <emb_f(x_g), emb_g(x_f)>
  const int r8 = raw[8], r9 = raw[9];
  #pragma unroll
  for (int f = 0; f < NUM_FIELDS; ++f) {
    #pragma unroll
    for (int g = f + 1; g < NUM_FIELDS; ++g) {
      float4 a0, a1, b0, b1;
      // left operand: table f, row in field-g range (LDS if g is 8 or 9)
      if (g == 8)
        load_row(&s8[(f * DIM8 + r8) * FACTOR_DIM], a0, a1);
      else if (g == 9)
        load_row(&s9[(f * DIM9 + r9) * FACTOR_DIM], a0, a1);
      else
        load_row(&emb[((size_t)f * INPUT_DIM + xo[g]) * FACTOR_DIM], a0, a1);
      // right operand: table g, row in field-f range (LDS only for pair (8,9))
      if (f == 8)
        load_row(&s8[(g * DIM8 + r8) * FACTOR_DIM], b0, b1);
      else
        load_row(&emb[((size_t)g * INPUT_DIM + xo[f]) * FACTOR_DIM], b0, b1);
      acc += dot8(a0, a1, b0, b1);
    }
  }
  out[b] = acc;
}

// ---------------- launcher ----------------
extern "C" void kernel_launch(void* const* d_in, const int* in_sizes, int n_in,
                              void* d_out, int out_size, void* d_ws, size_t ws_size,
                              hipStream_t stream) {
  const int*   x    = (const int*)d_in[0];
  const float* wlin = (const float*)d_in[1];
  const float* blin = (const float*)d_in[2];
  const float* emb  = (const float*)d_in[3];
  float*       out  = (float*)d_out;
  const int batch  = in_sizes[0] / NUM_FIELDS;      // 16384
  const int blocks = (batch + 255) / 256;           // 64
  FieldAwareFM_85633057947693_kernel<<<blocks, 256, 0, stream>>>(
      x, wlin, blin, emb, out, batch);
}